// freebase_bi_hGCN_21534966022329
// MI455X (gfx1250) — compile-verified
//
#include <hip/hip_runtime.h>

typedef float v2f __attribute__((ext_vector_type(2)));
typedef float v8f __attribute__((ext_vector_type(8)));

#define N_NODES 50000
#define N_PER   12500
#define NE      600000
#define HD      128

// ---------------------------------------------------------------------------
// C[M,128] = outscale[m] * maybe_relu( (A[M,K]*rowscale[m]) @ B[K,128] + bias[n] )
//
// V_WMMA_F32_16X16X4_F32, wave32. Each wave owns one N-tile (tn = wave&7),
// preloads ALL B fragments for its 16 columns into registers (K/4 x v2f),
// then strides over M-tiles reusing them: inner loop = 1 b64 A-load + 1 WMMA.
// Out-of-range A rows are CLAMPED to M-1 (they only feed never-stored D rows),
// so there is no EXEC divergence anywhere in the hot loop and EXEC is all-1s
// at every WMMA as required.
//
// A frag (16x4 f32): half=lane>>4: VGPR0 = K(k0+2*half), VGPR1 = K(k0+2*half+1)
// B frag (4x16 f32): same K striping, N = lane&15.
// D (16x16 f32): VGPR v -> row = v + 8*half, col = lane&15.
// ---------------------------------------------------------------------------
template <int K>
__global__ void gemm_wmma_f32(const float* __restrict__ A,
                              const float* __restrict__ B,
                              const float* __restrict__ bias,
                              const float* __restrict__ rowscale,  // nullable
                              const float* __restrict__ outscale,  // nullable
                              float* __restrict__ C,
                              int M, int relu)
{
    const int lane = threadIdx.x & 31;
    const int half = lane >> 4;     // 0 or 1
    const int lm   = lane & 15;
    const int wave   = (blockIdx.x * blockDim.x + threadIdx.x) >> 5;
    const int nwaves = (gridDim.x * blockDim.x) >> 5;   // multiple of 8
    const int tn  = wave & 7;                            // fixed N-tile
    const int col = tn * 16 + lm;

    // Preload B fragments for this wave's 16 columns (register-resident).
    v2f bf[K / 4];
    #pragma unroll
    for (int kk = 0; kk < K / 4; ++kk) {
        const int kb = kk * 4 + half * 2;
        bf[kk][0] = B[(long)kb * HD + col];
        bf[kk][1] = B[(long)(kb + 1) * HD + col];
    }

    const int mt = (M + 15) >> 4;
    const float bv = bias[col];

    for (int tm = wave >> 3; tm < mt; tm += (nwaves >> 3)) {
        const int row = tm * 16 + lm;
        const int rc  = (row < M) ? row : (M - 1);   // clamp: no divergence
        const float rs = (rowscale != nullptr) ? rowscale[rc] : 1.0f;
        const float* ap = A + (long)rc * K + half * 2;

        v8f acc = {};
        #pragma unroll
        for (int kk = 0; kk < K / 4; ++kk) {
            v2f a = *(const v2f*)(ap + kk * 4);
            a[0] *= rs;
            a[1] *= rs;
            acc = __builtin_amdgcn_wmma_f32_16x16x4_f32(
                false, a, false, bf[kk], (short)0, acc, false, false);
        }

        #pragma unroll
        for (int v = 0; v < 8; ++v) {
            const int orow = tm * 16 + v + half * 8;
            if (orow < M) {
                float val = acc[v] + bv;
                if (relu) val = fmaxf(val, 0.0f);
                if (outscale != nullptr) val *= outscale[orow];
                C[(long)orow * HD + col] = val;
            }
        }
    }
}

// ---------------------------------------------------------------------------
// Degree counts (atomic f32 add of 1.0) then in-place rsqrt(max(deg,1)).
// ---------------------------------------------------------------------------
__global__ void edge_degree(const int* __restrict__ src,
                            const int* __restrict__ dst,
                            float* __restrict__ degO,
                            float* __restrict__ degI)
{
    int i = blockIdx.x * blockDim.x + threadIdx.x;
    const int stride = gridDim.x * blockDim.x;
    for (; i < NE; i += stride) {
        atomicAdd(&degO[src[i]], 1.0f);
        atomicAdd(&degI[dst[i]], 1.0f);
    }
}

__global__ void make_norm(float* __restrict__ degO, float* __restrict__ degI)
{
    int i = blockIdx.x * blockDim.x + threadIdx.x;
    const int stride = gridDim.x * blockDim.x;
    for (; i < N_NODES; i += stride) {
        degO[i] = rsqrtf(fmaxf(degO[i], 1.0f));
        degI[i] = rsqrtf(fmaxf(degI[i], 1.0f));
    }
}

// ---------------------------------------------------------------------------
// accum[dst] += h[src]  (h already prescaled by out_norm).
// Each wave grabs 32 edges with ONE coalesced index load per array, then
// broadcasts via __shfl; per edge: coalesced 512B gather + 4 f32 atomics/lane
// into the L2-resident accumulator.  NE % 32 == 0 -> no tail.
// ---------------------------------------------------------------------------
__global__ void edge_aggregate(const float* __restrict__ h,
                               const int* __restrict__ src,
                               const int* __restrict__ dst,
                               float* __restrict__ accum)
{
    const int lane   = threadIdx.x & 31;
    const int wave   = (blockIdx.x * blockDim.x + threadIdx.x) >> 5;
    const int nwaves = (gridDim.x * blockDim.x) >> 5;
    for (long base = (long)wave * 32; base < NE; base += (long)nwaves * 32) {
        const int s = src[base + lane];
        const int d = dst[base + lane];
        for (int j = 0; j < 32; ++j) {
            const int ss = __shfl(s, j);
            const int dd = __shfl(d, j);
            const float4 v = *(const float4*)(h + (long)ss * HD + lane * 4);
            float* o = accum + (long)dd * HD + lane * 4;
            atomicAdd(o + 0, v.x);
            atomicAdd(o + 1, v.y);
            atomicAdd(o + 2, v.z);
            atomicAdd(o + 3, v.w);
        }
    }
}

// Layer-0 epilogue, in place:
//   h2[i] = relu(agg[i]*in_norm[row] + b_g0[col]) * out_norm[row]
// (out_norm prescale for the NEXT aggregate is fused here; relu>=0 so legal.)
__global__ void post_layer0(float* __restrict__ hb,
                            const float* __restrict__ in_norm,
                            const float* __restrict__ out_norm,
                            const float* __restrict__ bias)
{
    int i = blockIdx.x * blockDim.x + threadIdx.x;
    const int stride = gridDim.x * blockDim.x;
    const int total = N_NODES * HD;
    for (; i < total; i += stride) {
        const int row = i >> 7;
        const int col = i & (HD - 1);
        hb[i] = fmaxf(hb[i] * in_norm[row] + bias[col], 0.0f) * out_norm[row];
    }
}

// ---------------------------------------------------------------------------
// out[dst] += h[src]*(1 + 1{e in 0..4});  out[src] += h[dst]*1{e in {6,14,30}}
// Same 32-edge batched index loads; backward branch is wave-uniform.
// ---------------------------------------------------------------------------
__global__ void edge_final(const float* __restrict__ h,
                           const int* __restrict__ src,
                           const int* __restrict__ dst,
                           const int* __restrict__ ef,
                           float* __restrict__ out)
{
    const int lane   = threadIdx.x & 31;
    const int wave   = (blockIdx.x * blockDim.x + threadIdx.x) >> 5;
    const int nwaves = (gridDim.x * blockDim.x) >> 5;
    for (long base = (long)wave * 32; base < NE; base += (long)nwaves * 32) {
        const int s = src[base + lane];
        const int d = dst[base + lane];
        const int r = ef[base + lane];
        for (int j = 0; j < 32; ++j) {
            const int ss = __shfl(s, j);
            const int dd = __shfl(d, j);
            const int rr = __shfl(r, j);
            const float wf = (rr >= 0 && rr <= 4) ? 2.0f : 1.0f;

            const float4 hs = *(const float4*)(h + (long)ss * HD + lane * 4);
            float* od = out + (long)dd * HD + lane * 4;
            atomicAdd(od + 0, hs.x * wf);
            atomicAdd(od + 1, hs.y * wf);
            atomicAdd(od + 2, hs.z * wf);
            atomicAdd(od + 3, hs.w * wf);

            if (rr == 6 || rr == 14 || rr == 30) {      // wave-uniform
                const float4 hd = *(const float4*)(h + (long)dd * HD + lane * 4);
                float* os = out + (long)ss * HD + lane * 4;
                atomicAdd(os + 0, hd.x);
                atomicAdd(os + 1, hd.y);
                atomicAdd(os + 2, hd.z);
                atomicAdd(os + 3, hd.w);
            }
        }
    }
}

// ---------------------------------------------------------------------------
extern "C" void kernel_launch(void* const* d_in, const int* in_sizes, int n_in,
                              void* d_out, int out_size, void* d_ws, size_t ws_size,
                              hipStream_t stream)
{
    (void)in_sizes; (void)n_in; (void)out_size; (void)ws_size;

    const float* feat0 = (const float*)d_in[0];
    const float* Wfc0  = (const float*)d_in[1];
    const float* bfc0  = (const float*)d_in[2];
    const float* feat1 = (const float*)d_in[3];
    const float* Wfc1  = (const float*)d_in[4];
    const float* bfc1  = (const float*)d_in[5];
    const float* feat2 = (const float*)d_in[6];
    const float* Wfc2  = (const float*)d_in[7];
    const float* bfc2  = (const float*)d_in[8];
    const float* feat3 = (const float*)d_in[9];
    const float* Wfc3  = (const float*)d_in[10];
    const float* bfc3  = (const float*)d_in[11];
    const int*   src   = (const int*)d_in[12];
    const int*   dst   = (const int*)d_in[13];
    const int*   efeat = (const int*)d_in[14];
    const float* b_g0  = (const float*)d_in[15];
    const float* W_g1  = (const float*)d_in[16];
    const float* b_g1  = (const float*)d_in[17];
    float* out = (float*)d_out;

    const size_t NODE_ELEMS = (size_t)N_NODES * HD;    // 6.4M floats
    float* bufA = (float*)d_ws;                        // h / agg2
    float* bufB = bufA + NODE_ELEMS;                   // agg / h2 / h3
    float* degO = bufB + NODE_ELEMS;                   // -> out_norm
    float* degI = degO + N_NODES;                      // -> in_norm

    // --- degree norms (needed by the FC epilogues below) ---
    hipMemsetAsync(degO, 0, 2 * (size_t)N_NODES * sizeof(float), stream);
    edge_degree<<<1024, 256, 0, stream>>>(src, dst, degO, degI);
    make_norm<<<(N_NODES + 255) / 256, 256, 0, stream>>>(degO, degI);

    // --- per-type FC projections into h = bufA, prescaled by out_norm ---
    const int mt_per = (N_PER + 15) / 16;   // 782 blocks of 8 waves
    gemm_wmma_f32<128><<<mt_per, 256, 0, stream>>>(
        feat0, Wfc0, bfc0, nullptr, degO + 0 * N_PER,
        bufA + (size_t)0 * N_PER * HD, N_PER, 0);
    gemm_wmma_f32<128><<<mt_per, 256, 0, stream>>>(
        feat1, Wfc1, bfc1, nullptr, degO + 1 * N_PER,
        bufA + (size_t)1 * N_PER * HD, N_PER, 0);
    gemm_wmma_f32<64><<<mt_per, 256, 0, stream>>>(
        feat2, Wfc2, bfc2, nullptr, degO + 2 * N_PER,
        bufA + (size_t)2 * N_PER * HD, N_PER, 0);
    gemm_wmma_f32<64><<<mt_per, 256, 0, stream>>>(
        feat3, Wfc3, bfc3, nullptr, degO + 3 * N_PER,
        bufA + (size_t)3 * N_PER * HD, N_PER, 0);

    // --- GraphConv layer 0: aggregate; *in_norm,+b_g0,relu,*out_norm fused ---
    hipMemsetAsync(bufB, 0, NODE_ELEMS * sizeof(float), stream);
    edge_aggregate<<<2048, 256, 0, stream>>>(bufA, src, dst, bufB);
    post_layer0<<<(N_NODES * HD + 255) / 256, 256, 0, stream>>>(
        bufB, degI, degO, b_g0);

    // --- GraphConv layer 1: aggregate; (*in_norm) @ W_g1 + b_g1, relu ---
    hipMemsetAsync(bufA, 0, NODE_ELEMS * sizeof(float), stream);
    edge_aggregate<<<2048, 256, 0, stream>>>(bufB, src, dst, bufA);
    gemm_wmma_f32<128><<<3125, 256, 0, stream>>>(
        bufA, W_g1, b_g1, degI, nullptr, bufB, N_NODES, 1);

    // --- final bidirectional edge-typed aggregation ---
    hipMemsetAsync(out, 0, NODE_ELEMS * sizeof(float), stream);
    edge_final<<<2048, 256, 0, stream>>>(bufB, src, dst, efeat, out);
}